// IrregularDirectionalGradientConv_34144990003267
// MI455X (gfx1250) — compile-verified
//
#include <hip/hip_runtime.h>

// out[b,c,di,dj] = (1/961) * (A · X · A)[di,dj] - center_patch[di,dj]
// where A is the 31x31 symmetric banded-ones matrix A[r][c] = (|r-c| <= 15),
// X is one 31x31 plane, and center_patch[di,dj] = x_pad[ci+di, cj+dj].
// One block per (b,c) plane; 4 wave32s, each computing one 16x16 tile of the
// padded 32x32 result via V_WMMA_F32_16X16X4_F32.

typedef __attribute__((ext_vector_type(2))) float v2f;
typedef __attribute__((ext_vector_type(8))) float v8f;

#define KS   31
#define LTOT (KS * KS)      // 961
#define PADN 32
#define LSTR 33             // LDS row stride (avoid bank conflicts)

__global__ __launch_bounds__(128)
void idgc_wmma_kernel(const float* __restrict__ x,
                      const int* __restrict__ center_idx,
                      float* __restrict__ out) {
    __shared__ float sX[PADN * LSTR];   // plane, zero-padded to 32x32
    __shared__ float sA[PADN * LSTR];   // banded ones matrix, padded
    __shared__ float sT[PADN * LSTR];   // T = A * X

    const int plane = blockIdx.x;                 // 0..255 = b*C + c
    const float* xp = x + (size_t)plane * LTOT;
    float* op       = out + (size_t)plane * LTOT;

    const int tid = threadIdx.x;                  // 0..127

    // Stage X and build band matrix A (both padded with zeros to 32x32).
    for (int i = tid; i < PADN * PADN; i += 128) {
        const int r = i >> 5;
        const int c = i & 31;
        const bool in = (r < KS) && (c < KS);
        sX[r * LSTR + c] = in ? xp[r * KS + c] : 0.0f;
        const int d = r - c;
        sA[r * LSTR + c] = (in && d <= 15 && d >= -15) ? 1.0f : 0.0f;
    }
    __syncthreads();

    const int lane = tid & 31;
    const int wid  = tid >> 5;                    // wave id 0..3
    const int I    = (wid >> 1) << 4;             // tile row base: 0 or 16
    const int J    = (wid & 1) << 4;              // tile col base: 0 or 16
    const int half = lane >> 4;                   // 0 | 1
    const int ln16 = lane & 15;

    // ---- T = A * X (one 16x16 tile per wave, K swept 0..31 in steps of 4) ----
    v8f acc = {};
#pragma unroll
    for (int k0 = 0; k0 < PADN; k0 += 4) {
        const int ka = k0 + 2 * half;             // this lane's K base
        v2f a, b;
        a.x = sA[(I + ln16) * LSTR + (ka + 0)];
        a.y = sA[(I + ln16) * LSTR + (ka + 1)];
        b.x = sX[(ka + 0) * LSTR + (J + ln16)];
        b.y = sX[(ka + 1) * LSTR + (J + ln16)];
        acc = __builtin_amdgcn_wmma_f32_16x16x4_f32(
            /*neg_a=*/false, a, /*neg_b=*/false, b,
            /*c_mod=*/(short)0, acc, /*reuse_a=*/false, /*reuse_b=*/false);
    }
    // Scatter T tile to LDS (C/D layout: c[v] = D[v + 8*half][ln16]).
#pragma unroll
    for (int v = 0; v < 8; ++v) {
        sT[(I + v + 8 * half) * LSTR + (J + ln16)] = acc[v];
    }
    __syncthreads();

    // ---- S = T * A  (A is symmetric, so this is T * A^T as required) ----
    v8f acc2 = {};
#pragma unroll
    for (int k0 = 0; k0 < PADN; k0 += 4) {
        const int ka = k0 + 2 * half;
        v2f a, b;
        a.x = sT[(I + ln16) * LSTR + (ka + 0)];
        a.y = sT[(I + ln16) * LSTR + (ka + 1)];
        b.x = sA[(ka + 0) * LSTR + (J + ln16)];
        b.y = sA[(ka + 1) * LSTR + (J + ln16)];
        acc2 = __builtin_amdgcn_wmma_f32_16x16x4_f32(
            false, a, false, b, (short)0, acc2, false, false);
    }

    // ---- out = S/961 - center_patch ----
    const int cidx = *center_idx;
    const int ci = cidx / KS;
    const int cj = cidx % KS;
    const float inv = 1.0f / (float)LTOT;
#pragma unroll
    for (int v = 0; v < 8; ++v) {
        const int row = I + v + 8 * half;
        const int col = J + ln16;
        if (row < KS && col < KS) {
            // center_patch[row,col] = x_pad[ci+row, cj+col] = x[ci+row-15, cj+col-15] or 0
            const int xr = ci + row - 15;
            const int xc = cj + col - 15;
            const float cp = (xr >= 0 && xr < KS && xc >= 0 && xc < KS)
                                 ? sX[xr * LSTR + xc] : 0.0f;
            op[row * KS + col] = acc2[v] * inv - cp;
        }
    }
}

extern "C" void kernel_launch(void* const* d_in, const int* in_sizes, int n_in,
                              void* d_out, int out_size, void* d_ws, size_t ws_size,
                              hipStream_t stream) {
    (void)in_sizes; (void)n_in; (void)out_size; (void)d_ws; (void)ws_size;
    const float* x          = (const float*)d_in[0];
    const int*   center_idx = (const int*)d_in[1];
    float*       out        = (float*)d_out;

    // 256 planes (B*C = 8*32), one workgroup (4 wave32s) per plane.
    idgc_wmma_kernel<<<256, 128, 0, stream>>>(x, center_idx, out);
}